// enc_dec_attention_77738908057673
// MI455X (gfx1250) — compile-verified
//
#include <hip/hip_runtime.h>
#include <hip/hip_bf16.h>

typedef float v2f __attribute__((ext_vector_type(2)));
typedef float v8f __attribute__((ext_vector_type(8)));

__device__ __forceinline__ v8f wmma4(v2f a, v2f b, v8f c) {
  // V_WMMA_F32_16X16X4_F32: 8-arg pattern (neg_a, A, neg_b, B, c_mod, C, reuse_a, reuse_b)
  return __builtin_amdgcn_wmma_f32_16x16x4_f32(false, a, false, b, (short)0, c, false, false);
}

#define SC 3.36358566101486f  /* 128^0.25 */

// ---------------------------------------------------------------------------
// K0: fold Wk_pa / Wv_ma column groups (c -> 32-wide head dim) + biases
// ---------------------------------------------------------------------------
__global__ void fold_weights(const float* __restrict__ Wk_pa, const float* __restrict__ bk_pa,
                             const float* __restrict__ Wv_ma, const float* __restrict__ bv_ma,
                             float* __restrict__ Wkfold, float* __restrict__ Wvfold,
                             float* __restrict__ bkfold, float* __restrict__ bvfold) {
  int t = threadIdx.x;  // 128
  for (int d = 0; d < 32; ++d) {
    float a = 0.f, b = 0.f;
#pragma unroll
    for (int g4 = 0; g4 < 4; ++g4) {
      a += Wk_pa[t * 128 + g4 * 32 + d];
      b += Wv_ma[t * 128 + g4 * 32 + d];
    }
    Wkfold[t * 32 + d] = a;
    Wvfold[t * 32 + d] = b;
  }
  if (t < 32) {
    float a = 0.f, b = 0.f;
#pragma unroll
    for (int g4 = 0; g4 < 4; ++g4) { a += bk_pa[g4 * 32 + t]; b += bv_ma[g4 * 32 + t]; }
    bkfold[t] = 16384.f * a;  // 16384 rows per (b,h) quarter, each carries sum_g bias
    bvfold[t] = 16.f * b;     // 16 rows x 4 groups
  }
}

// ---------------------------------------------------------------------------
// K1: prot_adj column-sum, stage 1.  grid = 64 groups * 32 chunks, 256 thr.
// float4 loads: one wave reads one full 512B row per iteration (b128 path).
// Each block sums 512 rows x 128 cols.  Reads all 536 MB exactly once.
// ---------------------------------------------------------------------------
__global__ void prot_adj_partial(const float* __restrict__ pa, float* __restrict__ partial) {
  int blk = blockIdx.x;
  int g = blk >> 5;       // b*4+h
  int chunk = blk & 31;
  int lane = threadIdx.x & 31;
  int w = threadIdx.x >> 5;  // 0..7: row phase
  size_t base = ((size_t)g * 16384 + (size_t)chunk * 512) * 128;
  const float4* p = (const float4*)(pa + base) + (size_t)w * 32 + lane;
  float4 acc = {0.f, 0.f, 0.f, 0.f};
  for (int r = w; r < 512; r += 8) {
    float4 v = *p;
    acc.x += v.x; acc.y += v.y; acc.z += v.z; acc.w += v.w;
    p += 8 * 32;  // 8 rows
  }
  __shared__ float4 red[8][32];
  red[w][lane] = acc;
  __syncthreads();
  int t = threadIdx.x;
  if (t < 128) {           // column t = 4*l4 + j
    int l4 = t >> 2, j = t & 3;
    float s = 0.f;
#pragma unroll
    for (int ww = 0; ww < 8; ++ww) {
      const float* r4 = (const float*)&red[ww][l4];
      s += r4[j];
    }
    partial[blk * 128 + t] = s;
  }
}

// ---------------------------------------------------------------------------
// K2: finish S, compute ke_sum (LDS only), fold into per-(b,h) score weights:
//   Wtil[g][g4][c'] = sum_d Wq_ma[c'][g4*32+d] * ke_sum[g][d]
// grid 64, block 128
// ---------------------------------------------------------------------------
__global__ void finish_kesum_wtil(const float* __restrict__ partial,
                                  const float* __restrict__ Wkfold, const float* __restrict__ bkfold,
                                  const float* __restrict__ Wq_ma, const float* __restrict__ bq_ma,
                                  float* __restrict__ Wtil, float* __restrict__ btil) {
  int g = blockIdx.x;
  int t = threadIdx.x;
  __shared__ float S[128];
  __shared__ float ks[32];
  float s = 0.f;
  for (int ch = 0; ch < 32; ++ch) s += partial[(g * 32 + ch) * 128 + t];
  S[t] = s;
  __syncthreads();
  if (t < 32) {
    float acc = bkfold[t];
    for (int c = 0; c < 128; ++c) acc += S[c] * Wkfold[c * 32 + t];
    ks[t] = acc;
  }
  __syncthreads();
#pragma unroll
  for (int g4 = 0; g4 < 4; ++g4) {
    float acc = 0.f;
#pragma unroll
    for (int d = 0; d < 32; ++d) acc += Wq_ma[t * 128 + g4 * 32 + d] * ks[d];
    Wtil[(g * 4 + g4) * 128 + t] = acc;
  }
  if (t < 4) {
    float acc = 0.f;
#pragma unroll
    for (int d = 0; d < 32; ++d) acc += bq_ma[t * 32 + d] * ks[d];
    btil[g * 4 + t] = acc;
  }
}

// ---------------------------------------------------------------------------
// K3: fused mol_adj pass.  grid 4096 = (g=b*4+h)*64 + m, block 128.
// Loads the 16x128 tile of mol_adj backing (b,h,m); produces
//   scores_e[g][m][n] (n=0..63) and ve_sum[g][m][d] (d=0..31).
// ---------------------------------------------------------------------------
__global__ void mol_fused(const float* __restrict__ mol_adj,
                          const float* __restrict__ Wtil, const float* __restrict__ btil,
                          const float* __restrict__ Wvfold, const float* __restrict__ bvfold,
                          float* __restrict__ scores_e, float* __restrict__ vesum) {
  int blk = blockIdx.x;
  int g = blk >> 6;
  int m = blk & 63;
  int b = g >> 2, h = g & 3;
  int t = threadIdx.x;
  __shared__ float tile[16 * 128];
  __shared__ float T[128];
  __shared__ float wt[4 * 128];
  __shared__ float bt[4];
  __shared__ float wv[128 * 32];
  const float* src = mol_adj + ((size_t)b * 4096 + h * 1024 + m * 16) * 128;
  for (int i = t; i < 2048; i += 128) tile[i] = src[i];
  for (int i = t; i < 512; i += 128) wt[i] = Wtil[g * 512 + i];
  if (t < 4) bt[t] = btil[g * 4 + t];
  for (int i = t; i < 4096; i += 128) wv[i] = Wvfold[i];
  __syncthreads();
  float cs = 0.f;
#pragma unroll
  for (int r = 0; r < 16; ++r) cs += tile[r * 128 + t];
  T[t] = cs;
  __syncthreads();
  if (t < 64) {
    int n = t, rr = n >> 2, g4 = n & 3;
    float acc = bt[g4];
    const float* row = &tile[rr * 128];
    const float* w = &wt[g4 * 128];
    for (int c = 0; c < 128; ++c) acc += row[c] * w[c];
    scores_e[(g * 64 + m) * 64 + n] = SC * acc;
  } else if (t < 96) {
    int d = t - 64;
    float acc = bvfold[d];
    for (int c = 0; c < 128; ++c) acc += T[c] * wv[c * 32 + d];
    vesum[(g * 64 + m) * 32 + d] = acc;
  }
}

// ---------------------------------------------------------------------------
// K4: edge dual softmax (row over n + col over m of 64x64).  grid 64, blk 64.
// ---------------------------------------------------------------------------
__global__ void edge_message(const float* __restrict__ scores_e, float* __restrict__ msg_e) {
  int g = blockIdx.x, t = threadIdx.x;
  __shared__ float s[64 * 64];
  __shared__ float rmax[64], rsum[64], cmax[64], csum[64];
  for (int i = t; i < 4096; i += 64) s[i] = scores_e[g * 4096 + i];
  __syncthreads();
  float mx = -3.402823466e38f;
  for (int n = 0; n < 64; ++n) mx = fmaxf(mx, s[t * 64 + n]);
  float sm = 0.f;
  for (int n = 0; n < 64; ++n) sm += __expf(s[t * 64 + n] - mx);
  rmax[t] = mx; rsum[t] = sm;
  mx = -3.402823466e38f;
  for (int m = 0; m < 64; ++m) mx = fmaxf(mx, s[m * 64 + t]);
  sm = 0.f;
  for (int m = 0; m < 64; ++m) sm += __expf(s[m * 64 + t] - mx);
  cmax[t] = mx; csum[t] = sm;
  __syncthreads();
  for (int i = t; i < 4096; i += 64) {
    int m = i >> 6, n = i & 63;
    msg_e[g * 4096 + i] = __expf(s[i] - rmax[m]) / rsum[m] + __expf(s[i] - cmax[n]) / csum[n];
  }
}

// ---------------------------------------------------------------------------
// Generic WMMA f32 GEMM:  out[rows,128] = A[rows,128] @ W[128,128] + bias.
// W staged into LDS pair-interleaved: Wl2[kp*128+n] = {W[2kp][n], W[2kp+1][n]}
// so each B fragment is ONE aligned ds_load_b64 (no repacking movs).
// block = 256 (8 waves), each wave owns one 16-row strip; rows % 128 == 0.
// ---------------------------------------------------------------------------
__global__ __launch_bounds__(256) void gemm128_bias_wmma(
    const float* __restrict__ A, const float* __restrict__ W,
    const float* __restrict__ bias, float* __restrict__ out, int rows) {
  __shared__ v2f Wl2[64 * 128];  // 64 KB
  int t = threadIdx.x;
  for (int p = t; p < 8192; p += 256) {
    int kp = p >> 7, n = p & 127;
    v2f w;
    w.x = W[(2 * kp) * 128 + n];
    w.y = W[(2 * kp + 1) * 128 + n];
    Wl2[p] = w;
  }
  __syncthreads();
  int wv = t >> 5, lane = t & 31;
  int row0 = blockIdx.x * 128 + wv * 16;
  if (row0 >= rows) return;  // wave-uniform
  int lrow = lane & 15, khalf = lane >> 4;
  v8f acc[8] = {};
  const float* arow = A + (size_t)(row0 + lrow) * 128 + khalf * 2;
  for (int k0 = 0; k0 < 128; k0 += 4) {
    v2f a;
    a.x = arow[k0];
    a.y = arow[k0 + 1];
    int kp = (k0 >> 1) + khalf;  // pair index of column c1 = k0 + khalf*2
#pragma unroll
    for (int nt = 0; nt < 8; ++nt) {
      v2f bf = Wl2[kp * 128 + nt * 16 + lrow];
      acc[nt] = wmma4(a, bf, acc[nt]);
    }
  }
#pragma unroll
  for (int i = 0; i < 8; ++i) {
    int M = i + khalf * 8;
    float* orow = out + (size_t)(row0 + M) * 128;
#pragma unroll
    for (int nt = 0; nt < 8; ++nt)
      orow[nt * 16 + lrow] = acc[nt][i] + bias[nt * 16 + lrow];
  }
}

// ---------------------------------------------------------------------------
// K6a: node scores  s[g][m][n] = sc * q[b,h,m,:] . k[b,h,n,:]   grid 64, blk 256
// torch-view: q row = h*16+m/4 col (m%4)*32+d ;  k row = h*64+n/4 col (n%4)*32+d
// ---------------------------------------------------------------------------
__global__ void node_scores(const float* __restrict__ qproj, const float* __restrict__ kproj,
                            float* __restrict__ sout) {
  int g = blockIdx.x, t = threadIdx.x;
  int b = g >> 2, h = g & 3;
  __shared__ float qt[2048];
  const float* qp = qproj + (size_t)(b * 64 + h * 16) * 128;
  for (int i = t; i < 2048; i += 256) qt[i] = qp[i];
  __syncthreads();
  int n = t;
  const float* kr = kproj + (size_t)(b * 256 + h * 64 + (n >> 2)) * 128 + (n & 3) * 32;
  float kv[32];
#pragma unroll
  for (int d = 0; d < 32; ++d) kv[d] = kr[d];
  float* so = sout + (size_t)g * 16384;
  for (int m = 0; m < 64; ++m) {
    const float* qr = &qt[(m >> 2) * 128 + (m & 3) * 32];
    float acc = 0.f;
#pragma unroll
    for (int d = 0; d < 32; ++d) acc += qr[d] * kv[d];
    so[m * 256 + n] = SC * acc;
  }
}

// K6b: dual softmax in place (row over n=256, col over m=64) + row sums.
__global__ void node_message(float* __restrict__ msg, float* __restrict__ msgrs) {
  int g = blockIdx.x, t = threadIdx.x;
  __shared__ float rmax[64], rsum[64], cmax[256], csum[256];
  float* s = msg + (size_t)g * 16384;
  {
    float mx = -3.402823466e38f;
    for (int m = 0; m < 64; ++m) mx = fmaxf(mx, s[m * 256 + t]);
    float sm = 0.f;
    for (int m = 0; m < 64; ++m) sm += __expf(s[m * 256 + t] - mx);
    cmax[t] = mx; csum[t] = sm;
  }
  if (t < 64) {
    float mx = -3.402823466e38f;
    const float* row = s + t * 256;
    for (int n = 0; n < 256; ++n) mx = fmaxf(mx, row[n]);
    float sm = 0.f;
    for (int n = 0; n < 256; ++n) sm += __expf(row[n] - mx);
    rmax[t] = mx; rsum[t] = sm;
  }
  __syncthreads();
  if (t < 64) {
    float* row = s + t * 256;
    float rm = rmax[t], rs = rsum[t], tot = 0.f;
    for (int n = 0; n < 256; ++n) {
      float sv = row[n];
      float v = __expf(sv - rm) / rs + __expf(sv - cmax[n]) / csum[n];
      row[n] = v;
      tot += v;
    }
    msgrs[g * 64 + t] = tot;
  }
}

// K7: nh_flat[b][row][c] = msg_rowsum[b*4+row/16][(row%16)*4 + c/32] * vproj[b][row][c]
__global__ void build_nhflat(const float* __restrict__ vproj, const float* __restrict__ msgrs,
                             float* __restrict__ nhflat) {
  int idx = blockIdx.x * 256 + threadIdx.x;  // 131072 total
  int b = idx >> 13;
  int row = (idx >> 7) & 63;
  int c = idx & 127;
  int h = row >> 4;
  int m = ((row & 15) << 2) | (c >> 5);
  nhflat[idx] = msgrs[(b * 4 + h) * 64 + m] * vproj[idx];
}

// ---------------------------------------------------------------------------
// K9: edge output GEMM with on-the-fly A:
//   A[r=S*16+rr][g4*32+d] = msg_e[S][4*rr+g4] * vesum[S][d],  out = A@Wo_ma + bo
// W pair-interleaved in LDS as in gemm128_bias_wmma.
// grid 512 (8 strips/block), block 256.
// ---------------------------------------------------------------------------
__global__ __launch_bounds__(256) void edge_out_gemm(
    const float* __restrict__ msg_e, const float* __restrict__ vesum,
    const float* __restrict__ Wo_ma, const float* __restrict__ bo_ma,
    float* __restrict__ out) {
  __shared__ v2f Wl2[64 * 128];
  __shared__ float me[8][64];
  __shared__ float vs[8][32];
  int t = threadIdx.x;
  for (int p = t; p < 8192; p += 256) {
    int kp = p >> 7, n = p & 127;
    v2f w;
    w.x = Wo_ma[(2 * kp) * 128 + n];
    w.y = Wo_ma[(2 * kp + 1) * 128 + n];
    Wl2[p] = w;
  }
  int wv = t >> 5, lane = t & 31;
  int S = blockIdx.x * 8 + wv;  // strip id == g*64+m
  me[wv][lane] = msg_e[S * 64 + lane];
  me[wv][lane + 32] = msg_e[S * 64 + 32 + lane];
  vs[wv][lane & 31] = vesum[S * 32 + (lane & 31)];
  __syncthreads();
  int lrow = lane & 15, khalf = lane >> 4;
  v8f acc[8] = {};
  for (int k0 = 0; k0 < 128; k0 += 4) {
    int c1 = k0 + khalf * 2;
    v2f a;
    a.x = me[wv][4 * lrow + (c1 >> 5)] * vs[wv][c1 & 31];
    a.y = me[wv][4 * lrow + ((c1 + 1) >> 5)] * vs[wv][(c1 + 1) & 31];
    int kp = c1 >> 1;
#pragma unroll
    for (int nt = 0; nt < 8; ++nt) {
      v2f bf = Wl2[kp * 128 + nt * 16 + lrow];
      acc[nt] = wmma4(a, bf, acc[nt]);
    }
  }
  size_t rowbase = (size_t)S * 16;
#pragma unroll
  for (int i = 0; i < 8; ++i) {
    int M = i + khalf * 8;
    float* orow = out + (rowbase + M) * 128;
#pragma unroll
    for (int nt = 0; nt < 8; ++nt)
      orow[nt * 16 + lrow] = acc[nt][i] + bo_ma[nt * 16 + lrow];
  }
}

// ---------------------------------------------------------------------------
extern "C" void kernel_launch(void* const* d_in, const int* in_sizes, int n_in,
                              void* d_out, int out_size, void* d_ws, size_t ws_size,
                              hipStream_t stream) {
  (void)in_sizes; (void)n_in; (void)out_size; (void)ws_size;
  const float* mol_annot  = (const float*)d_in[0];
  const float* prot_annot = (const float*)d_in[1];
  const float* mol_adj    = (const float*)d_in[2];
  const float* prot_adj   = (const float*)d_in[3];
  const float* Wq_mx = (const float*)d_in[4];  const float* bq_mx = (const float*)d_in[5];
  const float* Wk_px = (const float*)d_in[6];  const float* bk_px = (const float*)d_in[7];
  const float* Wv_mx = (const float*)d_in[8];  const float* bv_mx = (const float*)d_in[9];
  const float* Wq_ma = (const float*)d_in[10]; const float* bq_ma = (const float*)d_in[11];
  const float* Wk_pa = (const float*)d_in[12]; const float* bk_pa = (const float*)d_in[13];
  const float* Wv_ma = (const float*)d_in[14]; const float* bv_ma = (const float*)d_in[15];
  const float* Wo_mx = (const float*)d_in[16]; const float* bo_mx = (const float*)d_in[17];
  const float* Wo_ma = (const float*)d_in[18]; const float* bo_ma = (const float*)d_in[19];

  float* out = (float*)d_out;
  float* out_edge = out;                      // 8,388,608
  float* out_node = out + 8388608;            // 131,072
  float* out_msg  = out + 8388608 + 131072;   // 1,048,576

  float* ws = (float*)d_ws;
  float* partial  = ws;                 // 262144
  float* Wkfold   = partial  + 262144;  // 4096
  float* Wvfold   = Wkfold   + 4096;    // 4096
  float* bkfold   = Wvfold   + 4096;    // 32
  float* bvfold   = bkfold   + 32;      // 32
  float* Wtil     = bvfold   + 32;      // 32768
  float* btil     = Wtil     + 32768;   // 256
  float* scores_e = btil     + 256;     // 262144
  float* msg_e    = scores_e + 262144;  // 262144
  float* vesum    = msg_e    + 262144;  // 131072
  float* qproj    = vesum    + 131072;  // 131072
  float* kproj    = qproj    + 131072;  // 524288
  float* vproj    = kproj    + 524288;  // 131072
  float* msgrs    = vproj    + 131072;  // 4096
  float* nhflat   = msgrs    + 4096;    // 131072   (total ~7.2 MB)

  // edge path
  fold_weights<<<1, 128, 0, stream>>>(Wk_pa, bk_pa, Wv_ma, bv_ma, Wkfold, Wvfold, bkfold, bvfold);
  prot_adj_partial<<<2048, 256, 0, stream>>>(prot_adj, partial);
  finish_kesum_wtil<<<64, 128, 0, stream>>>(partial, Wkfold, bkfold, Wq_ma, bq_ma, Wtil, btil);
  mol_fused<<<4096, 128, 0, stream>>>(mol_adj, Wtil, btil, Wvfold, bvfold, scores_e, vesum);
  edge_message<<<64, 64, 0, stream>>>(scores_e, msg_e);

  // node path projections (WMMA)
  gemm128_bias_wmma<<<8, 256, 0, stream>>>(mol_annot, Wq_mx, bq_mx, qproj, 1024);
  gemm128_bias_wmma<<<32, 256, 0, stream>>>(prot_annot, Wk_px, bk_px, kproj, 4096);
  gemm128_bias_wmma<<<8, 256, 0, stream>>>(mol_annot, Wv_mx, bv_mx, vproj, 1024);

  // node attention + message output
  node_scores<<<64, 256, 0, stream>>>(qproj, kproj, out_msg);
  node_message<<<64, 256, 0, stream>>>(out_msg, msgrs);
  build_nhflat<<<512, 256, 0, stream>>>(vproj, msgrs, nhflat);
  gemm128_bias_wmma<<<8, 256, 0, stream>>>(nhflat, Wo_mx, bo_mx, out_node, 1024);

  // edge output GEMM (WMMA, A built on the fly)
  edge_out_gemm<<<512, 256, 0, stream>>>(msg_e, vesum, Wo_ma, bo_ma, out_edge);
}